// GCN_14027363189187
// MI455X (gfx1250) — compile-verified
//
#include <hip/hip_runtime.h>

typedef __attribute__((ext_vector_type(2))) float v2f;
typedef __attribute__((ext_vector_type(8))) float v8f;

#define DIM 64

// ---------------- softmax over alpha (L+1 <= 8 values) ----------------
__global__ void gcn_softmax_kernel(const float* __restrict__ alpha,
                                   float* __restrict__ wts, int n) {
  if (blockIdx.x == 0 && threadIdx.x == 0) {
    float m = alpha[0];
    for (int i = 1; i < n; ++i) m = fmaxf(m, alpha[i]);
    float s = 0.f;
    for (int i = 0; i < n; ++i) s += expf(alpha[i] - m);
    for (int i = 0; i < n; ++i) wts[i] = expf(alpha[i] - m) / s;
  }
}

// ---------------- degree over destinations ----------------
__global__ void gcn_deg_kernel(const int* __restrict__ dst,
                               float* __restrict__ deg, int E) {
  int e = blockIdx.x * blockDim.x + threadIdx.x;
  if (e < E) unsafeAtomicAdd(&deg[dst[e]], 1.0f);
}

// ---------------- deg -> rsqrt(deg) (0 if deg==0), in place ----------------
__global__ void gcn_rsqrt_kernel(float* __restrict__ d, int n) {
  int i = blockIdx.x * blockDim.x + threadIdx.x;
  if (i < n) {
    float v = d[i];
    d[i] = (v > 0.f) ? rsqrtf(v) : 0.f;
  }
}

// ---------------- per-edge symmetric norm w = dis[src]*dis[dst] ----------------
__global__ void gcn_edgew_kernel(const int* __restrict__ src, const int* __restrict__ dst,
                                 const float* __restrict__ dis, float* __restrict__ w, int E) {
  int e = blockIdx.x * blockDim.x + threadIdx.x;
  if (e < E) w[e] = dis[src[e]] * dis[dst[e]];
}

// ---------------- acc = wts[0] * emb (float4 streams) ----------------
__global__ void gcn_scale_init_kernel(const float* __restrict__ x, float* __restrict__ o,
                                      const float* __restrict__ wts, int n4) {
  int i = blockIdx.x * blockDim.x + threadIdx.x;
  if (i < n4) {
    float s = wts[0];
    float4 a = ((const float4*)x)[i];
    float4 r;
    r.x = s * a.x; r.y = s * a.y; r.z = s * a.z; r.w = s * a.w;
    ((float4*)o)[i] = r;
  }
}

// ---------------- x_next[dst] += w[e] * x_cur[src]; 16 threads/edge, float4 ----------------
__global__ void gcn_spmm_kernel(const int* __restrict__ src, const int* __restrict__ dst,
                                const float* __restrict__ w, const float* __restrict__ xin,
                                float* __restrict__ xout, int E) {
  int t = blockIdx.x * blockDim.x + threadIdx.x;
  int e = t >> 4;
  if (e >= E) return;
  int c = (t & 15) << 2;
  int s = src[e];
  int d = dst[e];
  float we = w[e];
  const float4 xs = *(const float4*)(xin + (size_t)s * DIM + c);
  float* o = xout + (size_t)d * DIM + c;
  unsafeAtomicAdd(o + 0, we * xs.x);
  unsafeAtomicAdd(o + 1, we * xs.y);
  unsafeAtomicAdd(o + 2, we * xs.z);
  unsafeAtomicAdd(o + 3, we * xs.w);
}

// ---------------- acc += wts[k] * x (float4 streams) ----------------
__global__ void gcn_axpy_kernel(const float* __restrict__ x, float* __restrict__ o,
                                const float* __restrict__ wts, int k, int n4) {
  int i = blockIdx.x * blockDim.x + threadIdx.x;
  if (i < n4) {
    float s = wts[k];
    float4 a = ((const float4*)x)[i];
    float4 r = ((const float4*)o)[i];
    r.x += s * a.x; r.y += s * a.y; r.z += s * a.z; r.w += s * a.w;
    ((float4*)o)[i] = r;
  }
}

// ---------------- per-edge dot via WMMA: 16 edges/wave, diagonal of 16x16 ----------------
// A row m = acc[la[m]], B col n = acc[lb[n]].  f32 16x16x4 fragment layout:
//   lanes 0-15  : M/N = lane,    VGPR0/1 hold K = kk+0, kk+1
//   lanes 16-31 : M/N = lane-16, VGPR0/1 hold K = kk+2, kk+3
// -> each lane's A and B fragment is a float2 gather at row + kk + 2*(lane>>4).
// Diagonal of D: lanes 0-7 hold D[i][i] in c[lane]; lanes 24-31 hold D[i][i]
// (i = lane-16) in c[lane-24].
__global__ void gcn_edge_dot_wmma_kernel(const int* __restrict__ la, const int* __restrict__ lb,
                                         const float* __restrict__ x, float* __restrict__ res,
                                         int E) {
  const int gwave = (blockIdx.x * blockDim.x + threadIdx.x) >> 5;
  const int lane  = threadIdx.x & 31;
  const int ebase = gwave * 16;
  const int m     = lane & 15;

  int e = ebase + m;
  if (e >= E) e = E - 1;   // clamp (not predicate): EXEC must stay all-ones for WMMA

  const size_t ai = (size_t)la[e];
  const size_t bi = (size_t)lb[e];
  const int k0 = (lane >> 4) << 1;                 // 0 for lanes 0-15, 2 for lanes 16-31
  const float* ap = x + ai * DIM + k0;
  const float* bp = x + bi * DIM + k0;

  v8f c = {0.f, 0.f, 0.f, 0.f, 0.f, 0.f, 0.f, 0.f};
#pragma unroll
  for (int kk = 0; kk < DIM; kk += 4) {
    v2f a = *(const v2f*)(ap + kk);
    v2f b = *(const v2f*)(bp + kk);
    c = __builtin_amdgcn_wmma_f32_16x16x4_f32(false, a, false, b, (short)0, c, false, false);
  }

  // extract this lane's diagonal element
  const int r = lane & 7;
  float v = c[0];
  if (r == 1) v = c[1];
  if (r == 2) v = c[2];
  if (r == 3) v = c[3];
  if (r == 4) v = c[4];
  if (r == 5) v = c[5];
  if (r == 6) v = c[6];
  if (r == 7) v = c[7];

  const bool active = (lane < 8) || (lane >= 24);
  const int ei = ebase + ((lane < 8) ? lane : (lane - 16));
  if (active && ei < E) res[ei] = v;
}

extern "C" void kernel_launch(void* const* d_in, const int* in_sizes, int n_in,
                              void* d_out, int out_size, void* d_ws, size_t ws_size,
                              hipStream_t stream) {
  const int*   edge_index = (const int*)d_in[0];   // [2, E]
  const int*   eli        = (const int*)d_in[1];   // [2, E]
  const float* emb        = (const float*)d_in[2]; // [N, 64]
  const float* alpha      = (const float*)d_in[3]; // [L+1]

  const int E  = in_sizes[0] / 2;
  const int Nn = in_sizes[2] / DIM;
  const int L1 = in_sizes[3];      // L+1
  const int L  = L1 - 1;
  float* res = (float*)d_out;

  const int* src = edge_index;
  const int* dst = edge_index + E;
  const int* la  = eli;
  const int* lb  = eli + E;

  // ---- workspace carve-out (256B aligned blocks) ----
  size_t off = 0;
  auto carve = [&](size_t bytes) -> void* {
    void* p = (char*)d_ws + off;
    off = (off + bytes + 255) & ~(size_t)255;
    return p;
  };
  float* wts = (float*)carve(sizeof(float) * (size_t)L1);
  float* dis = (float*)carve(sizeof(float) * (size_t)Nn);
  float* w   = (float*)carve(sizeof(float) * (size_t)E);
  float* xA  = (float*)carve(sizeof(float) * (size_t)Nn * DIM);
  float* xB  = (float*)carve(sizeof(float) * (size_t)Nn * DIM);
  float* acc = (float*)carve(sizeof(float) * (size_t)Nn * DIM);
  (void)ws_size; (void)n_in; (void)out_size;

  const int n4 = Nn * DIM / 4;

  // 1. softmax(alpha)
  gcn_softmax_kernel<<<1, 1, 0, stream>>>(alpha, wts, L1);
  // 2. degree over dst (dis buffer doubles as deg)
  hipMemsetAsync(dis, 0, sizeof(float) * (size_t)Nn, stream);
  gcn_deg_kernel<<<(E + 255) / 256, 256, 0, stream>>>(dst, dis, E);
  gcn_rsqrt_kernel<<<(Nn + 255) / 256, 256, 0, stream>>>(dis, Nn);
  // 3. per-edge norm
  gcn_edgew_kernel<<<(E + 255) / 256, 256, 0, stream>>>(src, dst, dis, w, E);
  // 4. acc = wts[0] * emb
  gcn_scale_init_kernel<<<(n4 + 255) / 256, 256, 0, stream>>>(emb, acc, wts, n4);

  // 5. L propagation layers
  const float* xcur = emb;
  float* bufs[2] = {xA, xB};
  for (int l = 0; l < L; ++l) {
    float* xnext = bufs[l & 1];
    hipMemsetAsync(xnext, 0, sizeof(float) * (size_t)Nn * DIM, stream);
    long tcount = (long)E * (DIM / 4);
    gcn_spmm_kernel<<<(unsigned)((tcount + 255) / 256), 256, 0, stream>>>(
        src, dst, w, xcur, xnext, E);
    gcn_axpy_kernel<<<(n4 + 255) / 256, 256, 0, stream>>>(xnext, acc, wts, l + 1, n4);
    xcur = xnext;
  }

  // 6. per-edge dot products via WMMA (16 edges per wave, 8 waves per block)
  if (E > 0) {
    int waves  = (E + 15) / 16;
    int blocks = (waves + 7) / 8;
    gcn_edge_dot_wmma_kernel<<<blocks, 256, 0, stream>>>(la, lb, acc, res, E);
  }
}